// GraphAttentionV2Layer_12343736008997
// MI455X (gfx1250) — compile-verified
//
#include <hip/hip_runtime.h>
#include <hip/hip_bf16.h>
#include <math.h>

// GATv2 layer: N=1024 nodes, IN_F=128, HID=64, HEADS=4, neg_slope=0.2
// d_in: zh[N,128] f32, adj[N,N,1] i32, W_l[128,256] f32, W_r[128,256] f32, a[64] f32
// d_out: [N,64] f32
// d_ws layout (floats): Gl[1024*256] | Gr[1024*256] | att[4*1024*1024]  (= 18 MB)

typedef __attribute__((ext_vector_type(2))) float v2f;
typedef __attribute__((ext_vector_type(8))) float v8f;

#define NN     1024
#define INF_   128
#define HID_   64
#define HEADS_ 4
#define GCOLS  256   // HEADS*HID
#define ITILE  8     // i-rows per block in scores kernel

// ---------------------------------------------------------------------------
// Kernel 1: G = zh @ W  (fp32 WMMA 16x16x4).  One wave = one 16x16 tile.
// blockIdx.x: n-tile (0..15), blockIdx.y: m-tile (0..63), blockIdx.z: 0=W_l 1=W_r
// A-frag (16x4 f32): lanes 0-15 -> M=lane, VGPR v holds K = 2*(lane>=16)+v
// B-frag (4x16 f32): lanes 0-15 -> N=lane, VGPR v holds K = 2*(lane>=16)+v
// C/D (16x16 f32, 8 VGPRs): VGPR r -> M = r + 8*(lane>=16), N = lane%16
// ---------------------------------------------------------------------------
__global__ __launch_bounds__(32)
void gat_gemm_wmma(const float* __restrict__ zh,
                   const float* __restrict__ Wl,
                   const float* __restrict__ Wr,
                   float* __restrict__ Gl,
                   float* __restrict__ Gr)
{
    const int lane = threadIdx.x;
    const int lo = lane & 15;
    const int hi = lane >> 4;
    const int n0 = blockIdx.x * 16;     // output col within 256
    const int m0 = blockIdx.y * 16;     // node row
    const float* __restrict__ W = blockIdx.z ? Wr : Wl;
    float* __restrict__ G = blockIdx.z ? Gr : Gl;

    v8f c = {};
    for (int k0 = 0; k0 < INF_; k0 += 4) {
        v2f a = *(const v2f*)&zh[(m0 + lo) * INF_ + k0 + 2 * hi];
        v2f b;
        const int kr = k0 + 2 * hi;
        b.x = W[kr * GCOLS + n0 + lo];
        b.y = W[(kr + 1) * GCOLS + n0 + lo];
        c = __builtin_amdgcn_wmma_f32_16x16x4_f32(false, a, false, b,
                                                  (short)0, c, false, false);
    }
#pragma unroll
    for (int r = 0; r < 8; ++r) {
        G[(m0 + r + 8 * hi) * GCOLS + n0 + lo] = c[r];
    }
}

// ---------------------------------------------------------------------------
// Kernel 2: attention logits + masked softmax over j.
// One block per (i-tile of 8, h): 256 threads.
//   - g_r rows for the 8 i's cached in LDS; a[] cached in LDS.
//   - each thread owns 4 j's; g_l[j,:] loaded once into 64 registers and
//     reused against all 8 i rows  (cuts Gl L2 traffic 8x vs i-per-block).
//   - scores staged in LDS esm[8][1024]; 8 per-row block softmaxes.
// ---------------------------------------------------------------------------
__global__ __launch_bounds__(256)
void gat_scores_softmax(const float* __restrict__ Gl,
                        const float* __restrict__ Gr,
                        const int* __restrict__ adj,
                        const float* __restrict__ a,
                        float* __restrict__ att)
{
    const int i0 = blockIdx.x * ITILE;
    const int h  = blockIdx.y;
    const int t  = threadIdx.x;

    __shared__ float gr_s[ITILE][HID_];   // 2 KB
    __shared__ float av[HID_];            // 256 B
    __shared__ float esm[ITILE][NN];      // 32 KB
    __shared__ float red[256];            // 1 KB

    if (t < HID_) av[t] = a[t];
    for (int v = t; v < ITILE * HID_; v += 256) {
        const int il = v >> 6, f = v & (HID_ - 1);
        gr_s[il][f] = Gr[(i0 + il) * GCOLS + h * HID_ + f];
    }
    __syncthreads();

    for (int q = 0; q < 4; ++q) {
        const int j = t + q * 256;
        // g_l[j, h, :] -> registers
        float gl[HID_];
        const float* __restrict__ glp = &Gl[j * GCOLS + h * HID_];
#pragma unroll
        for (int f = 0; f < HID_; f += 4) {
            float4 g = *(const float4*)&glp[f];
            gl[f] = g.x; gl[f + 1] = g.y; gl[f + 2] = g.z; gl[f + 3] = g.w;
        }
        for (int il = 0; il < ITILE; ++il) {
            float s = 0.f;
#pragma unroll
            for (int f = 0; f < HID_; ++f) {
                const float x = gl[f] + gr_s[il][f];
                s += fmaxf(x, 0.2f * x) * av[f];   // leaky_relu(x,0.2)=max(x,0.2x)
            }
            esm[il][j] = (adj[(i0 + il) * NN + j] == 0) ? -INFINITY : s;
        }
    }
    __syncthreads();

    // per-row masked softmax over j
    for (int il = 0; il < ITILE; ++il) {
        float m = -INFINITY;
#pragma unroll
        for (int q = 0; q < 4; ++q) m = fmaxf(m, esm[il][t + q * 256]);
        red[t] = m;
        __syncthreads();
        for (int off = 128; off > 0; off >>= 1) {
            if (t < off) red[t] = fmaxf(red[t], red[t + off]);
            __syncthreads();
        }
        m = red[0];          // finite: self-loop guarantees >=1 valid entry
        __syncthreads();

        float p[4], ls = 0.f;
#pragma unroll
        for (int q = 0; q < 4; ++q) { p[q] = __expf(esm[il][t + q * 256] - m); ls += p[q]; }
        red[t] = ls;
        __syncthreads();
        for (int off = 128; off > 0; off >>= 1) {
            if (t < off) red[t] += red[t + off];
            __syncthreads();
        }
        const float inv = 1.f / red[0];
        __syncthreads();     // red[] reused next il

        float* __restrict__ row = &att[((size_t)h * NN + i0 + il) * NN];
#pragma unroll
        for (int q = 0; q < 4; ++q) row[t + q * 256] = p[q] * inv;
    }
}

// ---------------------------------------------------------------------------
// Kernel 3: out[i,f] = 0.25 * sum_h sum_j att[h][i][j] * Gr[j,h,f]
// 128-thread block = 4 waves; wave w computes f-tile w of the same 16-row
// m-tile, so the 4 waves share att fragments via the WGP cache.
// Accumulate across heads in the WMMA C regs.
// ---------------------------------------------------------------------------
__global__ __launch_bounds__(128)
void gat_aggregate_wmma(const float* __restrict__ att,
                        const float* __restrict__ Gr,
                        float* __restrict__ out)
{
    const int lane = threadIdx.x & 31;
    const int wave = threadIdx.x >> 5;  // 0..3 -> f-tile
    const int lo = lane & 15;
    const int hi = lane >> 4;
    const int f0 = wave * 16;           // 0..48
    const int m0 = blockIdx.x * 16;     // 0..1008

    v8f c = {};
    for (int h = 0; h < HEADS_; ++h) {
        const float* __restrict__ A = &att[(size_t)h * NN * NN];
        const float* __restrict__ B = &Gr[h * HID_ + f0 + lo];
        for (int j0 = 0; j0 < NN; j0 += 4) {
            // A: att_h[m0+lo, j0 + 2*hi + {0,1}]
            v2f a = *(const v2f*)&A[(m0 + lo) * NN + j0 + 2 * hi];
            // B: Gr[j0 + 2*hi + {0,1}, h*64 + f0 + lo]
            const int jr = j0 + 2 * hi;
            v2f b;
            b.x = B[jr * GCOLS];
            b.y = B[(jr + 1) * GCOLS];
            c = __builtin_amdgcn_wmma_f32_16x16x4_f32(false, a, false, b,
                                                      (short)0, c, false, false);
        }
    }
#pragma unroll
    for (int r = 0; r < 8; ++r) {
        out[(m0 + r + 8 * hi) * HID_ + f0 + lo] = c[r] * 0.25f;
    }
}

// ---------------------------------------------------------------------------
extern "C" void kernel_launch(void* const* d_in, const int* in_sizes, int n_in,
                              void* d_out, int out_size, void* d_ws, size_t ws_size,
                              hipStream_t stream)
{
    (void)in_sizes; (void)n_in; (void)out_size; (void)ws_size;

    const float* zh  = (const float*)d_in[0];
    const int*   adj = (const int*)  d_in[1];
    const float* Wl  = (const float*)d_in[2];
    const float* Wr  = (const float*)d_in[3];
    const float* a   = (const float*)d_in[4];
    float* out = (float*)d_out;

    float* Gl  = (float*)d_ws;                    // 1024*256 floats (1 MB)
    float* Gr  = Gl + (size_t)NN * GCOLS;         // 1024*256 floats (1 MB)
    float* att = Gr + (size_t)NN * GCOLS;         // 4*1024*1024 floats (16 MB)

    dim3 g1(GCOLS / 16, NN / 16, 2);              // (16, 64, 2)
    gat_gemm_wmma<<<g1, 32, 0, stream>>>(zh, Wl, Wr, Gl, Gr);

    dim3 g2(NN / ITILE, HEADS_);                  // (128, 4)
    gat_scores_softmax<<<g2, 256, 0, stream>>>(Gl, Gr, adj, a, att);

    gat_aggregate_wmma<<<NN / 16, 128, 0, stream>>>(att, Gr, out);
}